// Gate_27685359190355
// MI455X (gfx1250) — compile-verified
//
#include <hip/hip_runtime.h>

typedef float v2f __attribute__((ext_vector_type(2)));
typedef float v8f __attribute__((ext_vector_type(8)));

#define N_TOK   8192
#define DIM     4096
#define NEXP    64
#define BLOCK_M 32            // tokens per block (2 x 16-row WMMA subtiles)
#define KSEL    8

// One block = 32 tokens x 64 experts. 4 waves; wave w owns experts
// [16w, 16w+16) and accumulates TWO 16x16 tiles (token halves), so each
// B (weight) load feeds 2 WMMAs: load:WMMA = 1.5, B L2 traffic halved.
__global__ __launch_bounds__(128)
void gate_gemm_softmax_topk(const float* __restrict__ x,
                            const float* __restrict__ w,
                            float* __restrict__ topk_vals,
                            int*   __restrict__ topk_idx,
                            float* __restrict__ probs)
{
    __shared__ float s_scores[BLOCK_M][68];  // stride 68: no half-wave bank conflicts

    const int tid  = threadIdx.x;
    const int wave = tid >> 5;
    const int lane = tid & 31;
    const int half = lane >> 4;          // 0: K={k,k+1}, 1: K={k+2,k+3}
    const int lo   = lane & 15;

    const int m0 = blockIdx.x * BLOCK_M; // token base
    const int e0 = wave * 16;            // expert base for this wave

    // A: two token subtiles, row per lane, K-pair per half-wave.
    // B: w[e0+lo, :] (expert per lane == column of W^T, same K striping).
    const float* a0ptr = x + (size_t)(m0 + lo) * DIM + 2 * half;
    const float* a1ptr = a0ptr + (size_t)16 * DIM;
    const float* bptr  = w + (size_t)(e0 + lo) * DIM + 2 * half;

    v8f acc0 = {0.f, 0.f, 0.f, 0.f, 0.f, 0.f, 0.f, 0.f};
    v8f acc1 = {0.f, 0.f, 0.f, 0.f, 0.f, 0.f, 0.f, 0.f};

    #pragma unroll KSEL
    for (int k = 0; k < DIM; k += 4) {
        v2f b  = *(const v2f*)(bptr  + k);
        v2f a0 = *(const v2f*)(a0ptr + k);
        v2f a1 = *(const v2f*)(a1ptr + k);
        // D = A(16x4,f32) * B(4x16,f32) + C(16x16,f32); B reused twice.
        acc0 = __builtin_amdgcn_wmma_f32_16x16x4_f32(
                   false, a0, false, b, (short)0, acc0, false, false);
        acc1 = __builtin_amdgcn_wmma_f32_16x16x4_f32(
                   false, a1, false, b, (short)0, acc1, false, false);
    }

    // C/D layout: VGPR j of lane L holds M = j + 8*(L>>4), N = L&15
    #pragma unroll
    for (int j = 0; j < 8; ++j) {
        s_scores[j + 8 * half][e0 + lo]      = acc0[j];
        s_scores[16 + j + 8 * half][e0 + lo] = acc1[j];
    }
    __syncthreads();

    // One full wave: per-token softmax over 64 experts + top-8 selection.
    if (tid < BLOCK_M) {
        const int row = m0 + tid;

        float p[NEXP];
        float mx = -3.402823466e38f;
        #pragma unroll
        for (int e = 0; e < NEXP; ++e) {
            p[e] = s_scores[tid][e];
            mx = fmaxf(mx, p[e]);
        }
        float sum = 0.f;
        #pragma unroll
        for (int e = 0; e < NEXP; ++e) {
            p[e] = __expf(p[e] - mx);
            sum += p[e];
        }
        const float inv = 1.0f / sum;
        #pragma unroll
        for (int e = 0; e < NEXP; ++e) p[e] *= inv;

        // Vectorized probs writeback (global_store_b128).
        float4* probs4 = (float4*)(probs + (size_t)row * NEXP);
        #pragma unroll
        for (int i = 0; i < NEXP / 4; ++i) {
            probs4[i] = make_float4(p[4*i], p[4*i+1], p[4*i+2], p[4*i+3]);
        }

        // Top-8 by repeated argmax; strict '>' => lowest index wins ties,
        // matching jax.lax.top_k. 'taken' mask keeps p[] register-resident.
        unsigned long long taken = 0ull;
        float vals[8]; int idxs[8];
        float vsum = 0.f;
        #pragma unroll
        for (int kk = 0; kk < 8; ++kk) {
            float best = -1.0f; int bi = 0;
            #pragma unroll
            for (int e = 0; e < NEXP; ++e) {
                bool free_e = ((taken >> e) & 1ull) == 0ull;
                if (free_e && p[e] > best) { best = p[e]; bi = e; }
            }
            vals[kk] = best;
            idxs[kk] = bi;
            vsum += best;
            taken |= (1ull << bi);
        }
        const float rnorm = 1.0f / (vsum + 1e-9f);
        #pragma unroll
        for (int kk = 0; kk < 8; ++kk) {
            topk_vals[(size_t)row * 8 + kk] = vals[kk] * rnorm;
            topk_idx [(size_t)row * 8 + kk] = idxs[kk];
        }
    }
}

extern "C" void kernel_launch(void* const* d_in, const int* in_sizes, int n_in,
                              void* d_out, int out_size, void* d_ws, size_t ws_size,
                              hipStream_t stream) {
    const float* x = (const float*)d_in[0];   // [8192, 4096]
    const float* w = (const float*)d_in[1];   // [64, 4096]

    float* out   = (float*)d_out;
    float* vals  = out;                               // [8192, 8] f32
    int*   idx   = (int*)(out + (size_t)N_TOK * 8);   // [8192, 8] i32 (raw bits)
    float* probs = out + (size_t)2 * N_TOK * 8;       // [8192, 64] f32

    dim3 grid(N_TOK / BLOCK_M);   // 256 blocks
    dim3 block(128);
    hipLaunchKernelGGL(gate_gemm_softmax_topk, grid, block, 0, stream,
                       x, w, vals, idx, probs);
}